// CausalStructureLearner_28415503630420
// MI455X (gfx1250) — compile-verified
//
#include <hip/hip_runtime.h>
#include <hip/hip_bf16.h>

typedef __attribute__((ext_vector_type(2))) float v2f;
typedef __attribute__((ext_vector_type(8))) float v8f;

#define Bsz 32
#define Nn  256
#define Ff  256
#define Hh  64

// ---------------------------------------------------------------------------
// Kernel A1: Wcomb[f, c] (256 x 128), c<64 -> W_enc@W1[:64], c>=64 -> W_enc@W1[64:]
// ---------------------------------------------------------------------------
__global__ void build_wcomb(const float* __restrict__ W_enc,   // [256,64]
                            const float* __restrict__ W1,      // [128,64]
                            float* __restrict__ Wcomb) {       // [256,128]
    const int f = blockIdx.x;       // 0..255
    const int c = threadIdx.x;      // 0..127
    const int half = c >> 6;        // 0 or 1
    const int h = c & 63;
    const float* we = W_enc + f * Hh;
    const float* w1 = W1 + (half * Hh) * Hh + h;   // W1[(half*64 + k)*64 + h]
    float s = 0.0f;
#pragma unroll 8
    for (int k = 0; k < Hh; ++k)
        s = fmaf(we[k], w1[k * Hh], s);
    Wcomb[f * 128 + c] = s;
}

// ---------------------------------------------------------------------------
// Kernel A2: bias[h] = b1[h] + sum_k b_enc[k]*(W1[k,h] + W1[64+k,h])
// ---------------------------------------------------------------------------
__global__ void build_bias(const float* __restrict__ b_enc,
                           const float* __restrict__ W1,
                           const float* __restrict__ b1,
                           float* __restrict__ bias) {
    const int h = threadIdx.x;   // 0..63
    float s = b1[h];
#pragma unroll 8
    for (int k = 0; k < Hh; ++k)
        s = fmaf(b_enc[k], W1[k * Hh + h] + W1[(Hh + k) * Hh + h], s);
    bias[h] = s;
}

// ---------------------------------------------------------------------------
// Kernel B: fp32 WMMA GEMM  C[8192 x 128] = X[8192 x 256] @ Wcomb[256 x 128]
// Each wave: one 16-row strip x all 128 cols (8 C tiles), K-loop of 64 steps
// of V_WMMA_F32_16X16X4_F32. Cols 0..63 -> ai [B,N,H]; cols 64..127 -> aj
// stored TRANSPOSED as ajt [B,H,N] for coalesced reads in the score kernel.
// ---------------------------------------------------------------------------
__global__ void __launch_bounds__(128)
gemm_wmma_f32(const float* __restrict__ X,      // [8192, 256]
              const float* __restrict__ Wcomb,  // [256, 128]
              float* __restrict__ ai,           // [32*256, 64]
              float* __restrict__ ajt) {        // [32, 64, 256]
    const int lane    = threadIdx.x & 31;
    const int waveIdG = blockIdx.x * 4 + (threadIdx.x >> 5);  // 0..511
    const int rowBase = waveIdG * 16;                         // strip of 16 rows of X

    const int lrow  = lane & 15;        // row within tile (A) / col within tile (B,C)
    const int khalf = (lane >> 4) * 2;  // K sub-offset per fp32 WMMA layout

    v8f acc[8];
#pragma unroll
    for (int t = 0; t < 8; ++t) acc[t] = (v8f){0,0,0,0,0,0,0,0};

    const float* arow = X + (rowBase + lrow) * Ff;  // this lane's A row

    for (int kk = 0; kk < Ff; kk += 4) {
        const int kb = kk + khalf;
        // A fragment: A[lrow, kb], A[lrow, kb+1]  (aligned 8B load)
        v2f afrag = *(const v2f*)(arow + kb);
#pragma unroll
        for (int t = 0; t < 8; ++t) {
            const int col = t * 16 + lrow;
            v2f bfrag;
            bfrag.x = Wcomb[kb * 128 + col];
            bfrag.y = Wcomb[(kb + 1) * 128 + col];
            acc[t] = __builtin_amdgcn_wmma_f32_16x16x4_f32(
                false, afrag, false, bfrag, (short)0, acc[t], false, false);
        }
    }

    // Scatter C: vgpr v holds (m = v + 8*(lane>>4), n = lane&15) of each tile.
    const int mOff = 8 * (lane >> 4);
#pragma unroll
    for (int t = 0; t < 8; ++t) {
#pragma unroll
        for (int v = 0; v < 8; ++v) {
            const int m   = mOff + v;
            const int row = rowBase + m;       // global row in [0, 8192)
            const int n   = t * 16 + lrow;     // global col in [0, 128)
            const float val = acc[t][v];
            if (n < Hh) {
                ai[row * Hh + n] = val;                    // ai[b*N + i][h]
            } else {
                const int h = n - Hh;
                const int b = row >> 8;                    // row / 256
                const int j = row & 255;
                ajt[(b * Hh + h) * Nn + j] = val;          // ajt[b][h][j]
            }
        }
    }
}

// ---------------------------------------------------------------------------
// Kernel C: adjacency[b,i,j] = sigmoid( sum_h relu(ai[b,i,h]+ajt[b,h,j]+bias[h])*W2[h] + b2 )
//           with zeroed diagonal. One block per (b,i); thread j; ajt reads coalesced.
// ---------------------------------------------------------------------------
__global__ void __launch_bounds__(256)
score_kernel(const float* __restrict__ ai,     // [8192, 64]
             const float* __restrict__ ajt,    // [32, 64, 256]
             const float* __restrict__ bias,   // [64]
             const float* __restrict__ W2,     // [64]
             const float* __restrict__ b2,     // [1]
             float* __restrict__ adj) {        // [32, 256, 256]
    __shared__ float s_ai[Hh];
    __shared__ float s_w2[Hh];

    const int blk = blockIdx.x;          // 0..8191
    const int b   = blk >> 8;
    const int i   = blk & 255;
    const int j   = threadIdx.x;         // 0..255

    if (j < Hh) {
        s_ai[j] = ai[blk * Hh + j] + bias[j];
        s_w2[j] = W2[j];
    }
    __syncthreads();

    const float* ajrow = ajt + b * (Hh * Nn);
    float accv = 0.0f;
#pragma unroll 8
    for (int h = 0; h < Hh; ++h) {
        const float t = s_ai[h] + ajrow[h * Nn + j];
        accv = fmaf(s_w2[h], fmaxf(t, 0.0f), accv);
    }
    float val = 1.0f / (1.0f + __expf(-(accv + b2[0])));
    if (j == i) val = 0.0f;
    adj[blk * Nn + j] = val;
}

// ---------------------------------------------------------------------------
// Kernel D: structural = broadcast(structure_params) over batch
// ---------------------------------------------------------------------------
__global__ void broadcast_structural(const float* __restrict__ sp,   // [256,256]
                                     float* __restrict__ out2) {     // [32,256,256]
    const int idx = blockIdx.x * blockDim.x + threadIdx.x;  // 0..65535
    const float v = sp[idx];
#pragma unroll
    for (int b = 0; b < Bsz; ++b)
        out2[b * (Nn * Nn) + idx] = v;
}

// ---------------------------------------------------------------------------
extern "C" void kernel_launch(void* const* d_in, const int* in_sizes, int n_in,
                              void* d_out, int out_size, void* d_ws, size_t ws_size,
                              hipStream_t stream) {
    const float* X      = (const float*)d_in[0];  // [32,256,256]
    const float* W_enc  = (const float*)d_in[1];  // [256,64]
    const float* b_enc  = (const float*)d_in[2];  // [64]
    const float* W1     = (const float*)d_in[3];  // [128,64]
    const float* b1     = (const float*)d_in[4];  // [64]
    const float* W2     = (const float*)d_in[5];  // [64,1]
    const float* b2     = (const float*)d_in[6];  // [1]
    const float* sp     = (const float*)d_in[7];  // [256,256]

    float* out_adj    = (float*)d_out;                      // [32,256,256]
    float* out_struct = out_adj + Bsz * Nn * Nn;            // [32,256,256]

    float* ws    = (float*)d_ws;
    float* Wcomb = ws;                  // 256*128  = 32768 floats
    float* bias  = Wcomb + 256 * 128;   // 64 floats
    float* ai    = bias + 64;           // 8192*64  = 524288 floats
    float* ajt   = ai + 8192 * 64;      // 32*64*256 = 524288 floats

    // A: combined weights + fused bias
    build_wcomb<<<256, 128, 0, stream>>>(W_enc, W1, Wcomb);
    build_bias<<<1, 64, 0, stream>>>(b_enc, W1, b1, bias);

    // B: fp32 WMMA GEMM (512 waves, 4 waves/block)
    gemm_wmma_f32<<<128, 128, 0, stream>>>(X, Wcomb, ai, ajt);

    // C: fused relu-dot-sigmoid with zero diagonal
    score_kernel<<<Bsz * Nn, Nn, 0, stream>>>(ai, ajt, bias, W2, b2, out_adj);

    // D: structural broadcast
    broadcast_structural<<<(Nn * Nn) / 256, 256, 0, stream>>>(sp, out_struct);
}